// GMM_70042326663964
// MI455X (gfx1250) — compile-verified
//
#include <hip/hip_runtime.h>

// GMM background-model update (MOG, N=3) for MI455X / gfx1250.
// HBM-streaming workload: ~299 MB traffic/call -> ~12.8 us floor @ 23.3 TB/s.
// No matmul structure -> WMMA inapplicable. Optimized for minimum instruction
// overhead per byte: wave32, b128 global ops (36 vmem/thread, the minimum),
// CDNA5 NT cache policy on single-touch streams, 32-bit addressing, and
// single-instruction transcendentals (v_rcp/v_sqrt/v_exp) instead of IEEE
// divide sequences.

typedef float v4f __attribute__((ext_vector_type(4)));
typedef int   v4i __attribute__((ext_vector_type(4)));

#define ALPHA_C   0.2f
#define INVNORM_C 0.06349363593424097f    // 1 / (2*pi)^(3/2)
#define NLOG2E_H  -0.7213475204444817f    // -log2(e)/2 : exp(-t/2) = 2^(t*this)

__launch_bounds__(256)
__global__ void gmm_update_kernel(const float* __restrict__ frame,
                                  const float* __restrict__ mean,
                                  const float* __restrict__ devi,
                                  const float* __restrict__ wght,
                                  int*   __restrict__ out_mask,
                                  float* __restrict__ out_w,
                                  float* __restrict__ out_mean,
                                  float* __restrict__ out_dev,
                                  int nquad)               // nquad = H*W/4
{
  const int q = blockIdx.x * blockDim.x + threadIdx.x;
  if (q >= nquad) return;

  const v4f* f4 = reinterpret_cast<const v4f*>(frame);
  const v4f* m4 = reinterpret_cast<const v4f*>(mean);
  const v4f* d4 = reinterpret_cast<const v4f*>(devi);
  const v4f* w4 = reinterpret_cast<const v4f*>(wght);

  const int q3 = 3 * q;                     // all offsets stay 32-bit

  // ---- load 4 pixels of RGB frame data: 3 x b128 ----
  float f[12];
  {
    v4f a = f4[q3 + 0];
    v4f b = f4[q3 + 1];
    v4f c = f4[q3 + 2];
#pragma unroll
    for (int k = 0; k < 4; ++k) { f[k] = a[k]; f[4+k] = b[k]; f[8+k] = c[k]; }
  }

  // ---- pass 1: load all components, compute temp / distance mask / ratio ----
  float m[3][12];
  v4f dv[3], wv[3], tv[3], rv[3], rdv[3];
  unsigned dmask[3];

#pragma unroll
  for (int n = 0; n < 3; ++n) {
    const int mb = 3 * nquad * n + q3;      // < 2^27 in v4f units
    v4f a = m4[mb + 0];
    v4f b = m4[mb + 1];
    v4f c = m4[mb + 2];
#pragma unroll
    for (int k = 0; k < 4; ++k) { m[n][k] = a[k]; m[n][4+k] = b[k]; m[n][8+k] = c[k]; }

    // fully lane-coalesced single-touch streams -> non-temporal (CDNA5 TH=NT)
    dv[n] = __builtin_nontemporal_load(d4 + nquad * n + q);
    wv[n] = __builtin_nontemporal_load(w4 + nquad * n + q);

    unsigned bits = 0;
#pragma unroll
    for (int p = 0; p < 4; ++p) {
      const float dx = f[3*p+0] - m[n][3*p+0];
      const float dy = f[3*p+1] - m[n][3*p+1];
      const float dz = f[3*p+2] - m[n][3*p+2];
      const float sq = fmaf(dx, dx, fmaf(dy, dy, dz * dz));
      const float d  = dv[n][p];
      const float rd = __builtin_amdgcn_rcpf(d);      // v_rcp_f32, reused 3x
      const float t  = sq * rd * rd;                  // sqdist / dev^2
      tv[n][p]  = t;
      rdv[n][p] = rd;
      rv[n][p]  = wv[n][p] * rd;                      // weight / dev
      // sqrt(t) < 2.5*d  <=>  t < 6.25*d*d   (d > 0)
      if (t < 6.25f * (d * d)) bits |= (1u << p);
    }
    dmask[n] = bits;
  }

  const unsigned anybits = dmask[0] | dmask[1] | dmask[2];
  float rmax[4];
#pragma unroll
  for (int p = 0; p < 4; ++p)
    rmax[p] = fmaxf(rv[0][p], fmaxf(rv[1][p], rv[2][p]));

  unsigned showbits = 0;

  // ---- pass 2: EMA updates + streaming stores ----
#pragma unroll
  for (int n = 0; n < 3; ++n) {
    v4f nw, nd;
    float nm[12];
#pragma unroll
    for (int p = 0; p < 4; ++p) {
      const bool dm   = (dmask[n] >> p) & 1u;
      const bool some = (anybits   >> p) & 1u;
      const int  wum  = some ? (dm ? 1 : 0) : -1;
      const float d  = dv[n][p];
      const float w  = wv[n][p];
      const float t  = tv[n][p];
      const float rd = rdv[n][p];
      const bool  md = (rv[n][p] == rmax[p]);

      // exp(-t/2)/(NORM*d) : one v_exp_f32 + muls
      const float prob = __builtin_amdgcn_exp2f(t * NLOG2E_H) * (INVNORM_C * rd);
      const float rho  = ALPHA_C * prob;

      float w1 = w;
      if (wum == 1)      w1 = fmaf(1.0f - ALPHA_C, w, ALPHA_C);
      else if (wum == 0) w1 = (1.0f - ALPHA_C) * w;
      nw[p] = w1;

      float sq2 = 0.0f;
#pragma unroll
      for (int c = 0; c < 3; ++c) {
        const float mv = m[n][3*p+c];
        const float fv = f[3*p+c];
        float x = mv;
        if (wum == 1)       x = fmaf(rho, fv - mv, mv);   // (1-rho)*m + rho*f
        else if (wum == -1) x = fv;
        nm[3*p+c] = x;
        const float e = fv - x;
        sq2 = fmaf(e, e, sq2);
      }
      nd[p] = md ? fmaf(rho, sq2, __builtin_amdgcn_sqrtf(1.0f - rho) * d * d) : d;
      if (md && dm) showbits |= (1u << p);
    }

    __builtin_nontemporal_store(nw, reinterpret_cast<v4f*>(out_w)   + nquad * n + q);
    __builtin_nontemporal_store(nd, reinterpret_cast<v4f*>(out_dev) + nquad * n + q);
    const int mb = 3 * nquad * n + q3;
    v4f s0 = { nm[0], nm[1], nm[2],  nm[3]  };
    v4f s1 = { nm[4], nm[5], nm[6],  nm[7]  };
    v4f s2 = { nm[8], nm[9], nm[10], nm[11] };
    __builtin_nontemporal_store(s0, reinterpret_cast<v4f*>(out_mean) + mb + 0);
    __builtin_nontemporal_store(s1, reinterpret_cast<v4f*>(out_mean) + mb + 1);
    __builtin_nontemporal_store(s2, reinterpret_cast<v4f*>(out_mean) + mb + 2);
  }

  // ---- foreground mask: int32 0/255 replicated over 3 channels ----
  int ov[12];
#pragma unroll
  for (int p = 0; p < 4; ++p) {
    const int v = ((showbits >> p) & 1u) ? 255 : 0;
    ov[3*p+0] = v; ov[3*p+1] = v; ov[3*p+2] = v;
  }
  v4i o0 = { ov[0], ov[1], ov[2],  ov[3]  };
  v4i o1 = { ov[4], ov[5], ov[6],  ov[7]  };
  v4i o2 = { ov[8], ov[9], ov[10], ov[11] };
  __builtin_nontemporal_store(o0, reinterpret_cast<v4i*>(out_mask) + q3 + 0);
  __builtin_nontemporal_store(o1, reinterpret_cast<v4i*>(out_mask) + q3 + 1);
  __builtin_nontemporal_store(o2, reinterpret_cast<v4i*>(out_mask) + q3 + 2);
}

extern "C" void kernel_launch(void* const* d_in, const int* in_sizes, int n_in,
                              void* d_out, int out_size, void* d_ws, size_t ws_size,
                              hipStream_t stream)
{
  (void)n_in; (void)out_size; (void)d_ws; (void)ws_size;

  const float* frame = (const float*)d_in[0];   // (H,W,3)
  const float* mean  = (const float*)d_in[1];   // (N,H,W,3)
  const float* devi  = (const float*)d_in[2];   // (N,H,W)
  const float* wght  = (const float*)d_in[3];   // (N,H,W)

  const int HW    = in_sizes[0] / 3;            // H*W = 2,073,600 (multiple of 4)
  const int nquad = HW / 4;

  // outputs concatenated in return order: mask(int32 HW*3), weight(3*HW),
  // mean(3*HW*3), deviation(3*HW) -- all 32-bit elements, all 16B-aligned.
  int*   out_mask = (int*)d_out;
  float* out_w    = (float*)d_out + (size_t)HW * 3;
  float* out_mean = out_w    + (size_t)3 * HW;
  float* out_dev  = out_mean + (size_t)3 * HW * 3;

  dim3 block(256);                               // 8 wave32 per workgroup
  dim3 grid((nquad + 255) / 256);
  hipLaunchKernelGGL(gmm_update_kernel, grid, block, 0, stream,
                     frame, mean, devi, wght,
                     out_mask, out_w, out_mean, out_dev, nquad);
}